// BatchIrregularDownsample2d_8684423872931
// MI455X (gfx1250) — compile-verified
//
#include <hip/hip_runtime.h>

// BatchIrregularDownsample2d (D=2) for MI455X / gfx1250.
// Phase 1: per-batch index build (one 256-thread wave32-friendly block per batch).
// Phase 2: bandwidth-bound gather with non-temporal streaming loads/stores and
//          index prefetch (global_prefetch_b8). Zero FLOPs -> no WMMA by design.
// 3D grid removes all 64-bit div/mod from the hot gather kernel.

#define DLEV 2
#define BB   8
#define CC   256
#define HH   256
#define WW   256
#define SS   (HH / 2)        // 128  (subsampled grid side, stride-2 view)
#define PP   (SS * SS)       // 16384 subsampled positions per batch
#define SCAN_T 256
#define CHUNK  (PP / SCAN_T) // 64 positions per thread

static constexpr int CGRP = 8;   // channels handled per gather thread

// ---------------------------------------------------------------------------
// Phase 1: build gather index table G[B][M] (+ -1 padding sentinels).
// G[b] = [ 0..start-1 | start+rank(level==1 pos, row-major) | start+rank(level==2 topleft) | -1... ]
// ---------------------------------------------------------------------------
__global__ __launch_bounds__(SCAN_T)
void build_indices_kernel(const int* __restrict__ mask, int* __restrict__ G, int M) {
    __shared__ int s_ge1[SCAN_T];
    __shared__ int s_eq1[SCAN_T];
    __shared__ int s_tl [SCAN_T];
    __shared__ int s_tot[3];

    const int b = blockIdx.x;
    const int t = threadIdx.x;
    const int* __restrict__ m = mask + (size_t)b * (HH * WW);

    // Pass 1: local counts over this thread's contiguous chunk of the
    // row-major subsampled grid.
    int ge1 = 0, eq1 = 0, tl = 0;
    const int p0 = t * CHUNK;
    #pragma unroll 4
    for (int k = 0; k < CHUNK; ++k) {
        const int p = p0 + k;
        const int r = p >> 7;          // / SS
        const int c = p & (SS - 1);    // % SS
        const int v = m[(r << 1) * WW + (c << 1)];  // mask[::2, ::2]
        ge1 += (v >= 1);
        eq1 += (v == 1);
        tl  += (int)((v == 2) & ((r & 1) == 0) & ((c & 1) == 0));
    }
    s_ge1[t] = ge1; s_eq1[t] = eq1; s_tl[t] = tl;
    __syncthreads();

    // Exclusive scan of the 256 partials (serial by thread 0 -- tiny).
    if (t == 0) {
        int a = 0, e = 0, g2 = 0;
        for (int i = 0; i < SCAN_T; ++i) {
            int x = s_ge1[i]; s_ge1[i] = a;  a  += x;
            int y = s_eq1[i]; s_eq1[i] = e;  e  += y;
            int z = s_tl[i];  s_tl[i]  = g2; g2 += z;
        }
        s_tot[0] = a; s_tot[1] = e; s_tot[2] = g2;
    }
    __syncthreads();

    const int tot_ge1 = s_tot[0];
    const int n1      = s_tot[1];            // #level==1 subsampled positions
    const int n2      = s_tot[2];            // #level==2 top-left positions
    const int start   = 4 * (PP - tot_ge1);  // #full-res level-0 pixels
    const int total   = start + n1 + n2;     // valid length for this batch

    int* __restrict__ Gb = G + (size_t)b * M;

    // Pass 2: replay chunk with running ranks and scatter gather-indices.
    int rank = s_ge1[t];
    int o1   = s_eq1[t];
    int o2   = s_tl[t];
    for (int k = 0; k < CHUNK; ++k) {
        const int p = p0 + k;
        const int r = p >> 7;
        const int c = p & (SS - 1);
        const int v = m[(r << 1) * WW + (c << 1)];
        if (v >= 1) {
            if (v == 1) {
                Gb[start + o1] = start + rank;           // "dont_touch" entries
                ++o1;
            } else if (((r & 1) == 0) && ((c & 1) == 0)) {
                Gb[start + n1 + o2] = start + rank;      // pooled-down reps
                ++o2;
            }
            ++rank;
        }
    }

    // Identity prefix (untouched higher-res tokens).
    for (int j = t; j < start; j += SCAN_T) Gb[j] = j;
    // Padding sentinels (none for this setup's identical histograms, but general).
    for (int j = total + t; j < M; j += SCAN_T) Gb[j] = -1;
}

// ---------------------------------------------------------------------------
// Phase 2: streaming gather. 3D grid: x covers token index m (lanes ->
// consecutive m -> fully coalesced b32 traffic; M is odd so b128 vectors
// would be misaligned), y covers channel-groups, z covers batch. Each thread
// loads its gather index once into a register and reuses it across an
// unrolled CGRP=8 channel loop (8 independent outstanding loads per thread).
// Input/output are each touched exactly once -> non-temporal hints keep the
// 192MB L2 free for the hot index table; the index stream is prefetched.
// ---------------------------------------------------------------------------
__global__ __launch_bounds__(256)
void gather_kernel(const float* __restrict__ in, const int* __restrict__ G,
                   float* __restrict__ out, int N, int M) {
    const int m = blockIdx.x * 256 + threadIdx.x;
    if (m >= M) return;
    const int cg = blockIdx.y;
    const int b  = blockIdx.z;

    const int* __restrict__ Gp = G + (size_t)b * M + m;
    const int idx = *Gp;
    // Prefetch the next block-tile of indices (emits global_prefetch_b8;
    // speculative: a past-the-end address is silently dropped by HW).
    __builtin_prefetch(Gp + 256, 0, 1);

    const size_t row0 = ((size_t)b * CC + (size_t)cg * CGRP);
    const float* __restrict__ inb  = in  + row0 * N;
    float*       __restrict__ outb = out + row0 * M + m;

    if (idx >= 0) {
        const float* __restrict__ src = inb + idx;
        #pragma unroll
        for (int c = 0; c < CGRP; ++c) {
            const float v = __builtin_nontemporal_load(src + (size_t)c * N);
            __builtin_nontemporal_store(v, outb + (size_t)c * M);
        }
    } else {
        #pragma unroll
        for (int c = 0; c < CGRP; ++c)
            __builtin_nontemporal_store(0.0f, outb + (size_t)c * M);
    }
}

// ---------------------------------------------------------------------------
extern "C" void kernel_launch(void* const* d_in, const int* in_sizes, int n_in,
                              void* d_out, int out_size, void* d_ws, size_t ws_size,
                              hipStream_t stream) {
    const float* x    = (const float*)d_in[0];   // input  [B, C, N] f32
    const int*   mask = (const int*)  d_in[1];   // pooling_mask [B,1,H,W] i32
    float*       out  = (float*)d_out;           // output [B, C, M] f32

    const int BC = BB * CC;
    const int N  = in_sizes[0] / BC;             // 32776
    const int M  = out_size    / BC;             // 28681

    int* G = (int*)d_ws;                         // B*M ints (~918 KB) in scratch

    build_indices_kernel<<<BB, SCAN_T, 0, stream>>>(mask, G, M);

    const unsigned mblocks = (unsigned)((M + 255) / 256);
    dim3 grid(mblocks, CC / CGRP, BB);           // [113, 32, 8]
    gather_kernel<<<grid, 256, 0, stream>>>(x, G, out, N, M);
}